// KeyedConv2d_2413771620970
// MI455X (gfx1250) — compile-verified
//
#include <hip/hip_runtime.h>

typedef __attribute__((ext_vector_type(2))) float v2f;
typedef __attribute__((ext_vector_type(8))) float v8f;

// Conv as implicit GEMM on V_WMMA_F32_16X16X4_F32 (fp32 end-to-end):
//   M = out-channels (tile 16), N = flat pixels (two 16-wide tiles per wave),
//   K = kpos*64 + ci (576), stepped 4 at a time.
//
// SAME-padding is handled branch-free: addresses are clamped into the image
// and the loaded value is scaled by a {0,1} mask float, so the hot loop is
// 1 ds_load_b64 + 4 coalesced global_load_b32 + 4 v_mul + 2 WMMAs per step
// with no EXEC manipulation and no full-counter waits.

__global__ __launch_bounds__(256)
void KeyedConv2d_wmma_f32(const float* __restrict__ x,
                          const float* __restrict__ wt,
                          float* __restrict__ out) {
  // A-fragment-ordered weights: idx = ((kpos*16 + step)*32 + lane)*2 + t
  // holds Wt[co0 + lane%16][ci = step*4 + (lane/16)*2 + t][kpos]   (36 KB)
  __shared__ float ldsW[9 * 16 * 32 * 2];

  const int tid  = threadIdx.x;
  const int lane = tid & 31;
  const int wave = tid >> 5;

  const int coTile = blockIdx.x & 3;   // 4 co tiles of 16
  const int sGroup = blockIdx.x >> 2;  // 128 pixel groups of 256 px
  const int co0    = coTile * 16;

  // ---- cooperative weight staging into exact WMMA A-fragment order ----
  for (int idx = tid; idx < 9 * 16 * 32 * 2; idx += 256) {
    const int t    = idx & 1;
    const int l    = (idx >> 1) & 31;
    const int step = (idx >> 6) & 15;
    const int kpos = idx >> 10;              // 0..8
    const int m    = l & 15;
    const int kk   = ((l >> 4) << 1) + t;    // 0..3 within K-step
    const int ci   = step * 4 + kk;
    // Wt layout (OIHW): co*576 + ci*9 + kpos
    ldsW[idx] = wt[(co0 + m) * 576 + ci * 9 + kpos];
  }
  __syncthreads();

  // ---- per-wave tile: 32 consecutive output pixels of one image row ----
  const int pixBase = (sGroup * 8 + wave) * 32;  // flat (n,h,w)
  const int w0 = pixBase & 63;
  const int h  = (pixBase >> 6) & 63;
  const int n  = pixBase >> 12;

  const int mj   = lane & 15;          // A row m / B column j
  const int koff = (lane >> 4) << 1;   // lane's first K within a 4-step

  v8f acc0 = {0.f, 0.f, 0.f, 0.f, 0.f, 0.f, 0.f, 0.f};
  v8f acc1 = {0.f, 0.f, 0.f, 0.f, 0.f, 0.f, 0.f, 0.f};

  // base pointer for this image / this lane's starting channel
  const float* xn = x + n * 262144 + koff * 4096;

#pragma unroll
  for (int kpos = 0; kpos < 9; ++kpos) {
    const int dh = kpos / 3 - 1;   // compile-time constants after unroll
    const int dw = kpos % 3 - 1;

    // --- branch-free SAME padding: clamp address, mask value ---
    const int  hh  = h + dh;
    const int  hhc = hh < 0 ? 0 : (hh > 63 ? 63 : hh);
    const float fH = (hh == hhc) ? 1.0f : 0.0f;

    const int ww0 = w0 + mj + dw;                 // N-tile 0 column
    const int w0c = ww0 < 0 ? 0 : (ww0 > 63 ? 63 : ww0);
    const int ww1 = ww0 + 16;                     // N-tile 1 column (>= 15)
    const int w1c = ww1 > 63 ? 63 : ww1;

    const float f0 = (ww0 == w0c) ? fH : 0.0f;
    const float f1 = (ww1 == w1c) ? fH : 0.0f;

    const float* xp0 = xn + hhc * 64 + w0c;
    const float* xp1 = xn + hhc * 64 + w1c;
    const float* wp  = &ldsW[(kpos * 16) * 64 + lane * 2];

#pragma unroll
    for (int step = 0; step < 16; ++step) {
      v2f a;
      a.x = wp[0];
      a.y = wp[1];

      v2f b0, b1;
      b0.x = f0 * xp0[0];       // K = koff
      b0.y = f0 * xp0[4096];    // K = koff + 1
      b1.x = f1 * xp1[0];
      b1.y = f1 * xp1[4096];

      acc0 = __builtin_amdgcn_wmma_f32_16x16x4_f32(
          false, a, false, b0, (short)0, acc0, false, false);
      acc1 = __builtin_amdgcn_wmma_f32_16x16x4_f32(
          false, a, false, b1, (short)0, acc1, false, false);

      wp  += 64;          // next fragment (next 4 input channels)
      xp0 += 4 * 4096;    // advance 4 input channels
      xp1 += 4 * 4096;
    }
  }

  // ---- store D (VGPR r: lanes 0-15 -> M=r, lanes 16-31 -> M=r+8) ----
  const int jcol  = lane & 15;
  const int rofs  = (lane >> 4) << 3;
  const int obase = n * 262144 + h * 64;
#pragma unroll
  for (int r = 0; r < 8; ++r) {
    const int m = r + rofs;
    out[obase + (co0 + m) * 4096 + w0 + jcol]      = acc0[r];
    out[obase + (co0 + m) * 4096 + w0 + 16 + jcol] = acc1[r];
  }
}

extern "C" void kernel_launch(void* const* d_in, const int* in_sizes, int n_in,
                              void* d_out, int out_size, void* d_ws, size_t ws_size,
                              hipStream_t stream) {
  (void)in_sizes; (void)n_in; (void)out_size; (void)d_ws; (void)ws_size;
  const float* x  = (const float*)d_in[0];   // (8, 64, 64, 64) fp32
  const float* wt = (const float*)d_in[1];   // (64, 64, 3, 3)  fp32
  float* out      = (float*)d_out;           // (8, 64, 64, 64) fp32

  // 4 co-tiles * (32768 px / (8 waves * 32 px)) = 512 blocks of 256 threads
  dim3 grid(512), block(256);
  hipLaunchKernelGGL(KeyedConv2d_wmma_f32, grid, block, 0, stream, x, wt, out);
}